// AdaptiveLayer_51823075393961
// MI455X (gfx1250) — compile-verified
//
#include <hip/hip_runtime.h>
#include <hip/hip_bf16.h>
#include <stdint.h>

typedef __attribute__((ext_vector_type(2))) float    v2f;
typedef __attribute__((ext_vector_type(8))) float    v8f;
typedef __attribute__((ext_vector_type(4))) uint32_t v4u;
typedef __attribute__((ext_vector_type(8))) int      v8i;
typedef __attribute__((ext_vector_type(4))) int      v4i;

#define M_TOTAL  131072
#define N_DIM    128
#define K_DIM    512
#define BM       256                   // rows per block (8 waves x 32 rows)
#define KC       32
#define NKC      (K_DIM / KC)          // 16 K-chunks
#define A_STRIDE 36                    // 32 dwords + 4 pad dwords (TDM pad)
#define B_STRIDE 132                   // 128 dwords + 4 pad dwords (TDM pad)
#define A_FLOATS (BM * A_STRIDE)       // 9216
#define B_FLOATS (KC * B_STRIDE)       // 4224
#define BUF_FLOATS (A_FLOATS + B_FLOATS)

// ---------------------------------------------------------------------------
// TDM 2D tile load: builds D# group0 (128b) + group1 (256b) per
// CDNA5 ISA and issues TENSOR_LOAD_TO_LDS.
//   group0: count=1, lds_addr[63:32], global_addr[120:64], type=2 [127:126]
//   group1: data_size=2 (4B) [17:16], pad_enable [20], pad_interval [24:22],
//           pad_amount [31:25], tensor_dim0 [79:48], tensor_dim1 [111:80],
//           tile_dim0 [127:112], tile_dim1 [143:128],
//           tensor_dim0_stride [207:160]
// Toolchain uses the 6-arg builtin: (v4u, v8i, v4i, v4i, v8i, i32 cpol).
// ---------------------------------------------------------------------------
__device__ __forceinline__ void tdm_load_2d(uint32_t lds_off, const float* gptr,
                                            uint32_t tile_d0, uint32_t tile_d1,
                                            uint32_t tensor_d0, uint32_t tensor_d1,
                                            uint32_t stride0,
                                            uint32_t pad_interval, uint32_t pad_amount) {
  uint64_t ga = (uint64_t)(uintptr_t)gptr;
  v4u g0;
  g0.x = 1u;                                              // count=1, user mode
  g0.y = lds_off;                                         // lds_addr (bytes)
  g0.z = (uint32_t)ga;                                    // global_addr[31:0]
  g0.w = (uint32_t)((ga >> 32) & 0x01FFFFFFu) | (2u << 30); // addr[56:32] | type=2
  v8i g1;
  g1[0] = (int)((2u << 16) | (1u << 20) |                 // data_size=4B, pad_enable
                (pad_interval << 22) | (pad_amount << 25));
  g1[1] = (int)((tensor_d0 & 0xFFFFu) << 16);             // atomic_addr=0 | dim0 lo16
  g1[2] = (int)(((tensor_d0 >> 16) & 0xFFFFu) | ((tensor_d1 & 0xFFFFu) << 16));
  g1[3] = (int)(((tensor_d1 >> 16) & 0xFFFFu) | ((tile_d0 & 0xFFFFu) << 16));
  g1[4] = (int)(tile_d1 & 0xFFFFu);                       // tile_dim1 | tile_dim2=0
  g1[5] = (int)stride0;                                   // dim0_stride[31:0]
  g1[6] = 0;                                              // dim0_stride hi | dim1_stride lo
  g1[7] = 0;
  v4i z4; z4[0] = 0; z4[1] = 0; z4[2] = 0; z4[3] = 0;
  v8i z8; z8[0] = 0; z8[1] = 0; z8[2] = 0; z8[3] = 0;
  z8[4] = 0; z8[5] = 0; z8[6] = 0; z8[7] = 0;
  __builtin_amdgcn_tensor_load_to_lds(g0, g1, z4, z4, z8, 0);
}

// ---------------------------------------------------------------------------
// GEMM (M=131072, N=128, K=512) + bias + relu, fp32 WMMA 16x16x4.
// One block = 256 rows x 128 cols; 8 waves, each wave a 32x128 strip
// (2 M-subtiles x 8 N-subtiles = 16 accumulators); every B fragment from LDS
// feeds two WMMAs from registers. Double-buffered LDS chunks (KC=32) fed by
// the Tensor Data Mover.
// ---------------------------------------------------------------------------
__global__ __launch_bounds__(256, 1)
void gemm_bias_relu(const float* __restrict__ x, const float* __restrict__ W,
                    const float* __restrict__ bias, float* __restrict__ out) {
  __shared__ float lds[2 * BUF_FLOATS];

  const int tid  = threadIdx.x;
  const int wid  = tid >> 5;
  const int lane = tid & 31;
  const int half = lane >> 4;          // 0: K+0/1 (A) / M+0..7 (C); 1: K+2/3 / M+8..15
  const int l15  = lane & 15;
  const int blockRow = blockIdx.x * BM;

  v8f acc0[8] = {};
  v8f acc1[8] = {};

  // Prologue: pipeline 2 chunks deep. Wave0 -> A tiles, Wave1 -> B tiles.
  if (wid == 0) {
#pragma unroll
    for (int c = 0; c < 2; ++c) {
      uint32_t off = (uint32_t)(uintptr_t)&lds[c * BUF_FLOATS];
      tdm_load_2d(off, x + (size_t)blockRow * K_DIM + c * KC,
                  /*tile*/ KC, BM, /*tensor*/ K_DIM, 1u << 20,
                  /*stride*/ K_DIM, /*pad: every 32 dw*/ 4u, /*+4 dw*/ 3u);
    }
  } else if (wid == 1) {
#pragma unroll
    for (int c = 0; c < 2; ++c) {
      uint32_t off = (uint32_t)(uintptr_t)&lds[c * BUF_FLOATS + A_FLOATS];
      tdm_load_2d(off, W + (size_t)c * KC * N_DIM,
                  /*tile*/ N_DIM, KC, /*tensor*/ N_DIM, 1u << 20,
                  /*stride*/ N_DIM, /*pad: every 128 dw*/ 6u, /*+4 dw*/ 3u);
    }
  }

  const int rowA = wid * 32 + l15;     // first A-frag row for this lane

  for (int kc = 0; kc < NKC; ++kc) {
    const int buf = kc & 1;
    if (wid < 2) {                     // issuing waves gate on TENSORcnt
      if (kc < NKC - 1) __builtin_amdgcn_s_wait_tensorcnt(1);
      else              __builtin_amdgcn_s_wait_tensorcnt(0);
    }
    __syncthreads();                   // publish TDM data to all waves

    const float* As = &lds[buf * BUF_FLOATS];
    const float* Bs = &lds[buf * BUF_FLOATS + A_FLOATS];
#pragma unroll
    for (int kk = 0; kk < KC; kk += 4) {
      // A 16x4 f32 frags: lanes 0-15 hold K=kk,kk+1; lanes 16-31 K=kk+2,kk+3
      const float* ap0 = As + rowA * A_STRIDE + kk + 2 * half;
      const float* ap1 = ap0 + 16 * A_STRIDE;
      v2f a0; a0.x = ap0[0]; a0.y = ap0[1];
      v2f a1; a1.x = ap1[0]; a1.y = ap1[1];
#pragma unroll
      for (int nt = 0; nt < 8; ++nt) {
        // B 4x16 f32 frag: VGPR0 = K even rows, VGPR1 = K odd rows, N = l15
        const float* bp = Bs + (kk + 2 * half) * B_STRIDE + nt * 16 + l15;
        v2f bf; bf.x = bp[0]; bf.y = bp[B_STRIDE];
        acc0[nt] = __builtin_amdgcn_wmma_f32_16x16x4_f32(
            false, a0, false, bf, (short)0, acc0[nt], false, false);
        acc1[nt] = __builtin_amdgcn_wmma_f32_16x16x4_f32(
            false, a1, false, bf, (short)0, acc1[nt], false, false);
      }
    }
    __syncthreads();                   // everyone done with buf before refill
    if (kc + 2 < NKC) {
      if (wid == 0) {
        uint32_t off = (uint32_t)(uintptr_t)&lds[buf * BUF_FLOATS];
        tdm_load_2d(off, x + (size_t)blockRow * K_DIM + (kc + 2) * KC,
                    KC, BM, K_DIM, 1u << 20, K_DIM, 4u, 3u);
      } else if (wid == 1) {
        uint32_t off = (uint32_t)(uintptr_t)&lds[buf * BUF_FLOATS + A_FLOATS];
        tdm_load_2d(off, W + (size_t)(kc + 2) * KC * N_DIM,
                    N_DIM, KC, N_DIM, 1u << 20, N_DIM, 6u, 3u);
      }
    }
  }

  // Epilogue: C/D layout -> row = base + r + 8*half, col = nt*16 + l15
#pragma unroll
  for (int s = 0; s < 2; ++s) {
    const int rowBase = blockRow + wid * 32 + s * 16 + 8 * half;
#pragma unroll
    for (int nt = 0; nt < 8; ++nt) {
      const int col = nt * 16 + l15;
      const float bv = bias[col];
#pragma unroll
      for (int r = 0; r < 8; ++r) {
        float v = (s == 0 ? acc0[nt][r] : acc1[nt][r]) + bv;
        v = v > 0.0f ? v : 0.0f;
        out[(size_t)(rowBase + r) * N_DIM + col] = v;
      }
    }
  }
}

// ---------------------------------------------------------------------------
// MMD stage 1: one wave per pair p; reads 4 rows of `out` (L2-resident),
// wave-reduces 4 squared distances, applies 8 RBF kernels, block partial sum.
// ---------------------------------------------------------------------------
__global__ __launch_bounds__(256)
void mmd_partial(const float* __restrict__ out, float* __restrict__ partial) {
  __shared__ float sacc[8];
  const int wid  = threadIdx.x >> 5;
  const int lane = threadIdx.x & 31;
  const int p = blockIdx.x * 8 + wid;            // 4096 blocks * 8 waves = 32768 pairs

  const float* s0 = out + (size_t)(2 * p) * N_DIM;
  const float* s1 = out + (size_t)(2 * p + 1) * N_DIM;
  const float* t0 = out + (size_t)(M_TOTAL / 2 + 2 * p) * N_DIM;
  const float* t1 = out + (size_t)(M_TOTAL / 2 + 2 * p + 1) * N_DIM;

  float dss = 0.f, dtt = 0.f, dst = 0.f, dts = 0.f;
#pragma unroll
  for (int i = lane; i < N_DIM; i += 32) {
    float a = s0[i], b = s1[i], c = t0[i], d = t1[i];
    float e0 = a - b, e1 = c - d, e2 = a - d, e3 = b - c;
    dss += e0 * e0; dtt += e1 * e1; dst += e2 * e2; dts += e3 * e3;
  }
#pragma unroll
  for (int m = 16; m > 0; m >>= 1) {
    dss += __shfl_xor(dss, m, 32);
    dtt += __shfl_xor(dtt, m, 32);
    dst += __shfl_xor(dst, m, 32);
    dts += __shfl_xor(dts, m, 32);
  }
  if (lane == 0) {
    // gammas = (1/16) * 2^j, j=0..7  ->  inv_gamma = 16 * 0.5^j
    float g = 0.f, ig = 16.0f;
#pragma unroll
    for (int j = 0; j < 8; ++j) {
      g += __expf(-dss * ig) + __expf(-dtt * ig)
         - __expf(-dst * ig) - __expf(-dts * ig);
      ig *= 0.5f;
    }
    sacc[wid] = g * 0.125f;                      // mean over BETA=8
  }
  __syncthreads();
  if (threadIdx.x == 0) {
    float s = 0.f;
#pragma unroll
    for (int i = 0; i < 8; ++i) s += sacc[i];
    partial[blockIdx.x] = s;
  }
}

// MMD stage 2: deterministic final reduction of 4096 partials.
__global__ __launch_bounds__(256)
void mmd_final(const float* __restrict__ partial, float* __restrict__ loss) {
  __shared__ float red[256];
  float s = 0.f;
  for (int i = threadIdx.x; i < 4096; i += 256) s += partial[i];
  red[threadIdx.x] = s;
  __syncthreads();
  for (int k = 128; k > 0; k >>= 1) {
    if (threadIdx.x < k) red[threadIdx.x] += red[threadIdx.x + k];
    __syncthreads();
  }
  if (threadIdx.x == 0) loss[0] = red[0] * (2.0f / (float)(M_TOTAL / 2));
}

// ---------------------------------------------------------------------------
extern "C" void kernel_launch(void* const* d_in, const int* in_sizes, int n_in,
                              void* d_out, int out_size, void* d_ws, size_t ws_size,
                              hipStream_t stream) {
  const float* x = (const float*)d_in[0];
  const float* W = (const float*)d_in[1];
  const float* b = (const float*)d_in[2];
  float* out  = (float*)d_out;
  float* loss = out + (size_t)M_TOTAL * N_DIM;
  float* partial = (float*)d_ws;                 // 4096 floats of scratch

  gemm_bias_relu<<<M_TOTAL / BM, 256, 0, stream>>>(x, W, b, out);
  mmd_partial<<<4096, 256, 0, stream>>>(out, partial);
  mmd_final<<<1, 256, 0, stream>>>(partial, loss);
}